// TropicalCausalSelfAttention_12489764897084
// MI455X (gfx1250) — compile-verified
//
#include <hip/hip_runtime.h>
#include <hip/hip_bf16.h>
#include <math.h>

// ---------------------------------------------------------------------------
// Problem constants (B=1)
#define TT      512
#define NEMBD   768
#define NHEAD   12
#define NKV     6
#define HDIM    64
#define HALF    32

typedef float v2f __attribute__((ext_vector_type(2)));
typedef float v8f __attribute__((ext_vector_type(8)));

// ---------------------------------------------------------------------------
// Generic strided/batched fp32 GEMM:  C[b] = A[b] * B[b>>shift]
// Block = 128 threads (4 waves). Block tile: 64 (M) x 64 (N).
// Each wave owns a 64x16 slab: 4 independent V_WMMA_F32_16X16X4_F32
// accumulator chains reusing one B fragment -> 0.5 global loads per WMMA
// and 4-way ILP on the matrix pipe.
// A block-tile (64x64) is staged in LDS (pad 65: stride ≡ 1 mod 64 banks).
// Requires: M%64==0, N%64==0, K%64==0. All lanes active (EXEC all-ones).
__global__ __launch_bounds__(128) void wmma_gemm_f32_kernel(
    const float* __restrict__ A, const float* __restrict__ B,
    float* __restrict__ C,
    int K, int lda, int ldb, int ldc,
    long strideA, long strideB, long strideC, int bBatchShift)
{
    __shared__ float As[64][65];

    const int tid   = threadIdx.x;
    const int wave  = tid >> 5;
    const int lane  = tid & 31;
    const int lm    = lane & 15;          // lane within half-wave
    const int lh    = lane >> 4;          // half-wave select
    const int tileM = blockIdx.y;         // 64-row tile
    const int tileN = blockIdx.x * 4 + wave;
    const int batch = blockIdx.z;

    const float* Ab = A + (long)batch * strideA + (long)tileM * 64 * lda;
    const float* Bb = B + (long)(batch >> bBatchShift) * strideB;
    float*       Cb = C + (long)batch * strideC;
    const int    col = tileN * 16 + lm;

    v8f acc0 = {0.f,0.f,0.f,0.f,0.f,0.f,0.f,0.f};
    v8f acc1 = acc0, acc2 = acc0, acc3 = acc0;

    for (int k0 = 0; k0 < K; k0 += 64) {
        // Cooperatively stage A[64 x 64] into LDS (float4, fully coalesced).
        for (int i = tid; i < 1024; i += 128) {
            const int r  = i >> 4;
            const int c4 = (i & 15) << 2;
            const float4 v = *reinterpret_cast<const float4*>(Ab + (long)r * lda + k0 + c4);
            As[r][c4 + 0] = v.x;
            As[r][c4 + 1] = v.y;
            As[r][c4 + 2] = v.z;
            As[r][c4 + 3] = v.w;
        }
        __syncthreads();

        #pragma unroll
        for (int kk = 0; kk < 64; kk += 4) {
            const int kc = kk + 2 * lh;
            // One B fragment (4x16), reused by 4 WMMAs.
            v2f b;
            const float* Bp = Bb + (long)(k0 + kc) * ldb + col;
            b[0] = Bp[0];
            b[1] = Bp[ldb];
            // Four A fragments (16x4 each): VGPR0 = K0|K2, VGPR1 = K1|K3.
            v2f a0, a1, a2, a3;
            a0[0] = As[lm     ][kc]; a0[1] = As[lm     ][kc + 1];
            a1[0] = As[lm + 16][kc]; a1[1] = As[lm + 16][kc + 1];
            a2[0] = As[lm + 32][kc]; a2[1] = As[lm + 32][kc + 1];
            a3[0] = As[lm + 48][kc]; a3[1] = As[lm + 48][kc + 1];
            acc0 = __builtin_amdgcn_wmma_f32_16x16x4_f32(false, a0, false, b, (short)0, acc0, false, false);
            acc1 = __builtin_amdgcn_wmma_f32_16x16x4_f32(false, a1, false, b, (short)0, acc1, false, false);
            acc2 = __builtin_amdgcn_wmma_f32_16x16x4_f32(false, a2, false, b, (short)0, acc2, false, false);
            acc3 = __builtin_amdgcn_wmma_f32_16x16x4_f32(false, a3, false, b, (short)0, acc3, false, false);
        }
        __syncthreads();
    }

    // C/D layout: VGPR r -> row (r + 8*lane_half), col = lm.
    #pragma unroll
    for (int r = 0; r < 8; ++r) {
        const int row = tileM * 64 + r + 8 * lh;
        Cb[(long)(row     ) * ldc + col] = acc0[r];
        Cb[(long)(row + 16) * ldc + col] = acc1[r];
        Cb[(long)(row + 32) * ldc + col] = acc2[r];
        Cb[(long)(row + 48) * ldc + col] = acc3[r];
    }
}

// ---------------------------------------------------------------------------
// RoPE + RMS-norm, one wave32 per (token, head) row of HDIM=64.
// Lane i owns dims i and i+32 (the rotary pair). In-place.
__global__ void rope_rms_kernel(float* __restrict__ buf,
                                const float* __restrict__ cosb,
                                const float* __restrict__ sinb,
                                int nHeads)
{
    const int gwave = (int)((blockIdx.x * blockDim.x + threadIdx.x) >> 5);
    const int lane  = threadIdx.x & 31;
    if (gwave >= TT * nHeads) return;
    const int t = gwave / nHeads;
    const int h = gwave - t * nHeads;

    float* row = buf + ((long)t * nHeads + h) * HDIM;
    const float x1 = row[lane];
    const float x2 = row[lane + HALF];
    const float c  = cosb[t * HALF + lane];
    const float s  = sinb[t * HALF + lane];
    const float o1 = x1 * c - x2 * s;
    const float o2 = x2 * c + x1 * s;

    float ss = o1 * o1 + o2 * o2;
    #pragma unroll
    for (int off = 16; off > 0; off >>= 1)
        ss += __shfl_xor(ss, off, 32);

    const float scale = rsqrtf(ss * (1.0f / (float)HDIM) + 1e-6f);
    row[lane]        = o1 * scale;
    row[lane + HALF] = o2 * scale;
}

// ---------------------------------------------------------------------------
// Tropical (max-plus) scores + causal softmax, one 256-thread block per (q,h).
// Writes the normalized probability row (zeros above the diagonal) so the
// following P@V GEMM needs no masking. Max-plus has no tensor-core mapping;
// this is the irreducible VALU portion.
__global__ __launch_bounds__(256) void tropical_softmax_kernel(
    const float* __restrict__ Q, const float* __restrict__ Kb,
    float* __restrict__ P)
{
    const int qt  = blockIdx.x;     // query token
    const int h   = blockIdx.y;     // head
    const int kvh = h >> 1;         // GQA: 2 query heads per KV head
    const int tid = threadIdx.x;

    __shared__ float qrow[HDIM];
    __shared__ float red[256];

    if (tid < HDIM) qrow[tid] = Q[((long)qt * NHEAD + h) * HDIM + tid];
    __syncthreads();

    float s0 = -INFINITY, s1 = -INFINITY;
    {
        const int k = tid;
        if (k <= qt) {
            const float* kr = Kb + ((long)k * NKV + kvh) * HDIM;
            float s = -INFINITY;
            #pragma unroll
            for (int d = 0; d < HDIM; ++d) s = fmaxf(s, qrow[d] + kr[d]);
            s0 = s;
        }
    }
    {
        const int k = tid + 256;
        if (k <= qt) {
            const float* kr = Kb + ((long)k * NKV + kvh) * HDIM;
            float s = -INFINITY;
            #pragma unroll
            for (int d = 0; d < HDIM; ++d) s = fmaxf(s, qrow[d] + kr[d]);
            s1 = s;
        }
    }

    // Row max (stable softmax).
    red[tid] = fmaxf(s0, s1);
    __syncthreads();
    for (int off = 128; off > 0; off >>= 1) {
        if (tid < off) red[tid] = fmaxf(red[tid], red[tid + off]);
        __syncthreads();
    }
    const float m = red[0];
    __syncthreads();

    const float e0 = (tid       <= qt) ? __expf(s0 - m) : 0.f;
    const float e1 = (tid + 256 <= qt) ? __expf(s1 - m) : 0.f;

    red[tid] = e0 + e1;
    __syncthreads();
    for (int off = 128; off > 0; off >>= 1) {
        if (tid < off) red[tid] += red[tid + off];
        __syncthreads();
    }
    const float inv = 1.0f / red[0];

    float* prow = P + ((long)h * TT + qt) * TT;
    prow[tid]       = e0 * inv;
    prow[tid + 256] = e1 * inv;
}

// ---------------------------------------------------------------------------
extern "C" void kernel_launch(void* const* d_in, const int* in_sizes, int n_in,
                              void* d_out, int out_size, void* d_ws, size_t ws_size,
                              hipStream_t stream)
{
    (void)in_sizes; (void)n_in; (void)out_size; (void)ws_size;

    const float* x    = (const float*)d_in[0];   // [512][768]
    const float* cosb = (const float*)d_in[1];   // [512][32]
    const float* sinb = (const float*)d_in[2];   // [512][32]
    const float* Wq   = (const float*)d_in[3];   // [768][768]
    const float* Wk   = (const float*)d_in[4];   // [768][384]
    const float* Wv   = (const float*)d_in[5];   // [768][384]
    const float* Wp   = (const float*)d_in[6];   // [768][768]
    float*       out  = (float*)d_out;           // [512][768]

    // Workspace carve-up (floats): Q | K | V | P | Y   (~17 MB total)
    float* Qb = (float*)d_ws;                          // 512*768
    float* Kb = Qb + (long)TT * NEMBD;                 // 512*384
    float* Vb = Kb + (long)TT * NKV * HDIM;            // 512*384
    float* P  = Vb + (long)TT * NKV * HDIM;            // 12*512*512
    float* Yb = P  + (long)NHEAD * TT * TT;            // 512*768

    const dim3 blk(128);

    // 1) QKV projections (fp32 WMMA GEMMs).
    wmma_gemm_f32_kernel<<<dim3(NEMBD / 64, TT / 64, 1), blk, 0, stream>>>(
        x, Wq, Qb, NEMBD, NEMBD, NEMBD, NEMBD, 0, 0, 0, 0);
    wmma_gemm_f32_kernel<<<dim3((NKV * HDIM) / 64, TT / 64, 1), blk, 0, stream>>>(
        x, Wk, Kb, NEMBD, NEMBD, NKV * HDIM, NKV * HDIM, 0, 0, 0, 0);
    wmma_gemm_f32_kernel<<<dim3((NKV * HDIM) / 64, TT / 64, 1), blk, 0, stream>>>(
        x, Wv, Vb, NEMBD, NEMBD, NKV * HDIM, NKV * HDIM, 0, 0, 0, 0);

    // 2) RoPE + RMS-norm on Q and K.
    rope_rms_kernel<<<(TT * NHEAD * 32) / 128, blk, 0, stream>>>(Qb, cosb, sinb, NHEAD);
    rope_rms_kernel<<<(TT * NKV   * 32) / 128, blk, 0, stream>>>(Kb, cosb, sinb, NKV);

    // 3) Tropical max-plus scores + causal softmax (VALU path).
    tropical_softmax_kernel<<<dim3(TT, NHEAD, 1), dim3(256), 0, stream>>>(Qb, Kb, P);

    // 4) y_h = P_h @ V_{h>>1}  (batched over 12 heads; GQA via batch>>1).
    //    A: P   lda=512, strideA=512*512
    //    B: V   ldb=384, strideB=64 applied to (head>>1)
    //    C: Y   ldc=768, strideC=64 (head h owns columns h*64..h*64+63)
    wmma_gemm_f32_kernel<<<dim3(HDIM / 64, TT / 64, NHEAD), blk, 0, stream>>>(
        P, Vb, Yb, TT, TT, NKV * HDIM, NEMBD,
        (long)TT * TT, (long)HDIM, (long)HDIM, 1);

    // 5) Output projection.
    wmma_gemm_f32_kernel<<<dim3(NEMBD / 64, TT / 64, 1), blk, 0, stream>>>(
        Yb, Wp, out, NEMBD, NEMBD, NEMBD, NEMBD, 0, 0, 0, 0);
}